// UserModel_33114197852551
// MI455X (gfx1250) — compile-verified
//
#include <hip/hip_runtime.h>
#include <hip/hip_bf16.h>

// ---------------------------------------------------------------------------
// Problem constants (from reference)
// ---------------------------------------------------------------------------
#define BB      64
#define TT      200
#define NUM_C1  2000
#define NUM_C2  200
#define NUM_DD  10000
#define DV      128

typedef __attribute__((ext_vector_type(2))) float v2f;
typedef __attribute__((ext_vector_type(8))) float v8f;

// workspace layout (floats)
#define WS_UD      0                     // [384]
#define WS_UR      384                   // [384]
#define WS_U1      768                   // [128]
#define WS_U2      896                   // [128]
#define WS_U3      1024                  // [128]
#define WS_U4      1152                  // [128]
#define WS_C1TAB   1280                  // [64*2000]
#define WS_C2TAB   (WS_C1TAB + BB*NUM_C1)        // [64*200]
#define WS_NEWC1   (WS_C2TAB + BB*NUM_C2)        // [T*64]
#define WS_NEWC2   (WS_NEWC1 + TT*BB)            // [T*64]
#define WS_TOTAL   (WS_NEWC2 + TT*BB)

// output layout (floats), concat in reference return order
#define OUT_ALPHA  0
#define OUT_H      (BB*TT)                           // 12800
#define OUT_C1     (OUT_H + BB*TT*DV)                // 1651200
#define OUT_C2     (OUT_C1 + (long)BB*TT*NUM_C1)     // 27251200

__device__ __forceinline__ float sigmoidf_(float x) {
    return 1.0f / (1.0f + __expf(-x));
}

// ---------------------------------------------------------------------------
// K0: zero the C1/C2 state tables in workspace (ws is poisoned by harness)
// ---------------------------------------------------------------------------
__global__ void k_zero_tables(float* ws) {
    int n = BB * NUM_C1 + BB * NUM_C2;
    for (int i = blockIdx.x * blockDim.x + threadIdx.x; i < n;
         i += gridDim.x * blockDim.x)
        ws[WS_C1TAB + i] = 0.0f;
}

// ---------------------------------------------------------------------------
// K1: rank-collapse precompute.
//   u_d[j] = sum_k v_d[k] * w_ih[j, k]        (k in [0,128))
//   u_r[j] = sum_k v_r[k] * w_ih[j, 128+k]
//   u1..u4[n] = v_c1/v_c2/v_d/v_r contracted against W2a row blocks
// ---------------------------------------------------------------------------
__global__ void k_precompute(const float* __restrict__ v_c1,
                             const float* __restrict__ v_c2,
                             const float* __restrict__ v_d,
                             const float* __restrict__ v_r,
                             const float* __restrict__ w_ih,   // [384,256]
                             const float* __restrict__ W2a,    // [512,128]
                             float* ws) {
    int tid = threadIdx.x;
    if (tid < 3 * DV) {
        float sd = 0.0f, sr = 0.0f;
        const float* row = w_ih + (long)tid * (2 * DV);
        for (int k = 0; k < DV; ++k) {
            sd += v_d[k] * row[k];
            sr += v_r[k] * row[DV + k];
        }
        ws[WS_UD + tid] = sd;
        ws[WS_UR + tid] = sr;
    }
    if (tid < DV) {
        float s1 = 0.f, s2 = 0.f, s3 = 0.f, s4 = 0.f;
        for (int k = 0; k < DV; ++k) {
            s1 += v_c1[k] * W2a[(long)(k          ) * DV + tid];
            s2 += v_c2[k] * W2a[(long)(k +     DV) * DV + tid];
            s3 += v_d [k] * W2a[(long)(k + 2 * DV) * DV + tid];
            s4 += v_r [k] * W2a[(long)(k + 3 * DV) * DV + tid];
        }
        ws[WS_U1 + tid] = s1;
        ws[WS_U2 + tid] = s2;
        ws[WS_U3 + tid] = s3;
        ws[WS_U4 + tid] = s4;
    }
}

// ---------------------------------------------------------------------------
// K2: GRU scan. Batch rows are independent through the recurrence, so split
// the batch over 4 workgroups (16 rows each). 256 threads = 8 waves; wave jb
// owns hidden cols [jb*16, +16) for all three gates. At 2 waves/SIMD the
// loop-invariant B-fragments of w_hh (3 gates x 32 k-steps x v2f = 192
// floats/lane) live in REGISTERS across the whole 200-step scan -> no
// global/scratch traffic for w_hh inside the recurrence. h lives in LDS.
// ---------------------------------------------------------------------------
__global__ __launch_bounds__(256, 1)
void k_gru(const int*   __restrict__ d_seq,   // [B,T]
           const float* __restrict__ r_seq,   // [B,T]
           const float* __restrict__ Dtab,    // [NUM_D]
           const float* __restrict__ w_hh,    // [384,128]
           const float* __restrict__ b_ih,    // [384]
           const float* __restrict__ b_hh,    // [384]
           const float* __restrict__ ws,      // u_d/u_r
           float* __restrict__ h_out) {       // [B,T,DV]
    __shared__ float h_lds[16][DV + 4];       // pad: conflict-free col reads
    __shared__ float gamma_s[16];
    __shared__ float rr_s[16];

    const int tid  = threadIdx.x;
    const int lane = tid & 31;
    const int jb   = tid >> 5;                // wave = hidden 16-col block
    const int half = lane >> 4;               // 0/1
    const int lm   = lane & 15;               // 0..15
    const int mb   = blockIdx.x;              // batch 16-row block (0..3)
    const int j    = jb * 16 + lm;            // this lane's hidden column

    // ---- loop-invariant per-lane gate constants (12 regs) ----
    const float ud0 = ws[WS_UD + j];
    const float ud1 = ws[WS_UD + DV + j];
    const float ud2 = ws[WS_UD + 2 * DV + j];
    const float ur0 = ws[WS_UR + j];
    const float ur1 = ws[WS_UR + DV + j];
    const float ur2 = ws[WS_UR + 2 * DV + j];
    const float bi0 = b_ih[j], bi1 = b_ih[DV + j], bi2 = b_ih[2 * DV + j];
    const float bh0 = b_hh[j], bh1 = b_hh[DV + j], bh2 = b_hh[2 * DV + j];

    // ---- preload B fragments for the whole K=128 reduction, 3 gates ----
    // B[k][n] = w_hh[n][k]; lane holds n = nbase+lm, vgpr pair holds k0+{0,1}
    v2f Bf[3][32];
    #pragma unroll
    for (int g = 0; g < 3; ++g) {
        const long nrow = (long)(g * DV + jb * 16 + lm) * DV;
        #pragma unroll
        for (int kk = 0; kk < 32; ++kk) {
            const int k0 = kk * 4 + 2 * half;
            Bf[g][kk].x = w_hh[nrow + k0];
            Bf[g][kk].y = w_hh[nrow + k0 + 1];
        }
    }

    // h0 = 0
    for (int i = tid; i < 16 * (DV + 4); i += 256)
        (&h_lds[0][0])[i] = 0.0f;
    __syncthreads();

    for (int t = 0; t < TT; ++t) {
        if (tid < 16) {
            const int b = mb * 16 + tid;
            gamma_s[tid] = Dtab[d_seq[b * TT + t]];
            rr_s[tid]    = r_seq[b * TT + t];
        }
        __syncthreads();

        v8f cr = {}, cz = {}, cn = {};
        #pragma unroll
        for (int kk = 0; kk < 32; ++kk) {
            v2f a;
            a.x = h_lds[lm][kk * 4 + 2 * half];
            a.y = h_lds[lm][kk * 4 + 2 * half + 1];
            cr = __builtin_amdgcn_wmma_f32_16x16x4_f32(false, a, false, Bf[0][kk],
                                                       (short)0, cr, false, false);
            cz = __builtin_amdgcn_wmma_f32_16x16x4_f32(false, a, false, Bf[1][kk],
                                                       (short)0, cz, false, false);
            cn = __builtin_amdgcn_wmma_f32_16x16x4_f32(false, a, false, Bf[2][kk],
                                                       (short)0, cn, false, false);
        }

        // gate math in C/D layout: vgpr v -> local row v + 8*half, col = lm
        float hnew[8];
        #pragma unroll
        for (int v = 0; v < 8; ++v) {
            const int m = v + 8 * half;             // local batch row 0..15
            const float g  = gamma_s[m];
            const float rv = rr_s[m];
            const float r = sigmoidf_(g * ud0 + rv * ur0 + bi0 + cr[v] + bh0);
            const float z = sigmoidf_(g * ud1 + rv * ur1 + bi1 + cz[v] + bh1);
            const float n = tanhf(g * ud2 + rv * ur2 + bi2 + r * (cn[v] + bh2));
            hnew[v] = (1.0f - z) * n + z * h_lds[m][j];
        }
        __syncthreads();   // all reads of h done
        #pragma unroll
        for (int v = 0; v < 8; ++v) {
            const int m = v + 8 * half;
            h_lds[m][j] = hnew[v];
            h_out[(long)(mb * 16 + m) * TT * DV + (long)t * DV + j] = hnew[v];
        }
        __syncthreads();   // writes visible before next step
    }
}

// ---------------------------------------------------------------------------
// K3: alpha = relu(h @ W1a + b1a) @ W1b + b1b, fused.
// Grid: 800 blocks = M-tiles over the 12800 (b,t) rows; 8 waves/block,
// wave w owns N-tile w. fp32 WMMA, then shuffle-reduce the W1b contraction.
// ---------------------------------------------------------------------------
__global__ __launch_bounds__(256)
void k_alpha(const float* __restrict__ h_seq,  // [B*T, DV]
             const float* __restrict__ W1a,    // [DV, DV]
             const float* __restrict__ b1a,    // [DV]
             const float* __restrict__ W1b,    // [DV, 1]
             const float* __restrict__ b1b,    // [1]
             float* __restrict__ alpha) {      // [B*T]
    __shared__ float part[16][8];
    const int tid  = threadIdx.x;
    const int lane = tid & 31;
    const int wave = tid >> 5;                 // N-tile 0..7
    const int half = lane >> 4;
    const int lm   = lane & 15;
    const int mt   = blockIdx.x;               // M-tile
    const int row  = mt * 16 + lm;
    const int nb   = wave * 16;

    v8f c = {};
    for (int k0 = 0; k0 < DV; k0 += 4) {
        v2f a, b;
        a.x = h_seq[(long)row * DV + k0 + 2 * half];
        a.y = h_seq[(long)row * DV + k0 + 2 * half + 1];
        b.x = W1a[(long)(k0 + 2 * half) * DV + nb + lm];
        b.y = W1a[(long)(k0 + 2 * half + 1) * DV + nb + lm];
        c = __builtin_amdgcn_wmma_f32_16x16x4_f32(false, a, false, b,
                                                  (short)0, c, false, false);
    }
    #pragma unroll
    for (int v = 0; v < 8; ++v) {
        const int j = nb + lm;
        float val = fmaxf(c[v] + b1a[j], 0.0f) * W1b[j];
        // reduce over the 16 cols of this row (lanes within each half)
        val += __shfl_xor(val, 1);
        val += __shfl_xor(val, 2);
        val += __shfl_xor(val, 4);
        val += __shfl_xor(val, 8);
        if (lm == 0) part[v + 8 * half][wave] = val;
    }
    __syncthreads();
    if (tid < 16) {
        float s = b1b[0];
        #pragma unroll
        for (int w = 0; w < 8; ++w) s += part[tid][w];
        alpha[mt * 16 + tid] = s;
    }
}

// ---------------------------------------------------------------------------
// K4: C1/C2 scan. Each batch row only touches its own table rows, so the 64
// scans are independent: 64 threads, each serial over T. The [B,512]@W2a
// GEMM is rank-4 collapsed to 4 scalars * precomputed vectors.
// ---------------------------------------------------------------------------
__global__ __launch_bounds__(64)
void k_cscan(const int*   __restrict__ c1_seq,
             const int*   __restrict__ c2_seq,
             const int*   __restrict__ d_seq,
             const float* __restrict__ r_seq,
             const float* __restrict__ Dtab,
             const float* __restrict__ b2a,    // [DV]
             const float* __restrict__ W2b,    // [DV,2]
             const float* __restrict__ b2b,    // [2]
             float* ws) {
    const int b = threadIdx.x;
    float* C1t = ws + WS_C1TAB + (long)b * NUM_C1;
    float* C2t = ws + WS_C2TAB + (long)b * NUM_C2;
    const float* u1 = ws + WS_U1;
    const float* u2 = ws + WS_U2;
    const float* u3 = ws + WS_U3;
    const float* u4 = ws + WS_U4;

    for (int t = 0; t < TT; ++t) {
        const int   c1 = c1_seq[b * TT + t];
        const int   c2 = c2_seq[b * TT + t];
        const float g  = Dtab[d_seq[b * TT + t]];
        const float rv = r_seq[b * TT + t];
        const float beta1 = C1t[c1];
        const float beta2 = C2t[c2];
        float o0 = b2b[0], o1 = b2b[1];
        for (int n = 0; n < DV; ++n) {
            const float pre = fmaxf(
                beta1 * u1[n] + beta2 * u2[n] + g * u3[n] + rv * u4[n] + b2a[n],
                0.0f);
            o0 += pre * W2b[n * 2 + 0];
            o1 += pre * W2b[n * 2 + 1];
        }
        C1t[c1] = o0;
        C2t[c2] = o1;
        ws[WS_NEWC1 + t * BB + b] = o0;
        ws[WS_NEWC2 + t * BB + b] = o1;
    }
}

// ---------------------------------------------------------------------------
// K5/K6: reconstruct the full snapshot histories (the HBM-roofline part:
// ~105 MB of coalesced stores). Thread (b,j) replays the T updates of its
// table cell; stores along j are contiguous.
// ---------------------------------------------------------------------------
__global__ void k_fill_c1(const int* __restrict__ c1_seq,
                          const float* __restrict__ ws,
                          float* __restrict__ out) {  // [B,T,NUM_C1]
    const int b = blockIdx.y;
    const int j = blockIdx.x * blockDim.x + threadIdx.x;
    if (j >= NUM_C1) return;
    float cur = 0.0f;
    for (int t = 0; t < TT; ++t) {
        if (c1_seq[b * TT + t] == j) cur = ws[WS_NEWC1 + t * BB + b];
        out[(long)b * TT * NUM_C1 + (long)t * NUM_C1 + j] = cur;
    }
}

__global__ void k_fill_c2(const int* __restrict__ c2_seq,
                          const float* __restrict__ ws,
                          float* __restrict__ out) {  // [B,T,NUM_C2]
    const int b = blockIdx.y;
    const int j = blockIdx.x * blockDim.x + threadIdx.x;
    if (j >= NUM_C2) return;
    float cur = 0.0f;
    for (int t = 0; t < TT; ++t) {
        if (c2_seq[b * TT + t] == j) cur = ws[WS_NEWC2 + t * BB + b];
        out[(long)b * TT * NUM_C2 + (long)t * NUM_C2 + j] = cur;
    }
}

// ---------------------------------------------------------------------------
extern "C" void kernel_launch(void* const* d_in, const int* in_sizes, int n_in,
                              void* d_out, int out_size, void* d_ws, size_t ws_size,
                              hipStream_t stream) {
    const int*   c1_seq = (const int*)  d_in[0];
    const int*   c2_seq = (const int*)  d_in[1];
    const int*   d_seq  = (const int*)  d_in[2];
    const float* r_seq  = (const float*)d_in[3];
    const float* Dtab   = (const float*)d_in[4];
    const float* v_c1   = (const float*)d_in[5];
    const float* v_c2   = (const float*)d_in[6];
    const float* v_d    = (const float*)d_in[7];
    const float* v_r    = (const float*)d_in[8];
    const float* w_ih   = (const float*)d_in[9];
    const float* w_hh   = (const float*)d_in[10];
    const float* b_ih   = (const float*)d_in[11];
    const float* b_hh   = (const float*)d_in[12];
    const float* W1a    = (const float*)d_in[13];
    const float* b1a    = (const float*)d_in[14];
    const float* W1b    = (const float*)d_in[15];
    const float* b1b    = (const float*)d_in[16];
    const float* W2a    = (const float*)d_in[17];
    const float* b2a    = (const float*)d_in[18];
    const float* W2b    = (const float*)d_in[19];
    const float* b2b    = (const float*)d_in[20];

    float* out = (float*)d_out;
    float* ws  = (float*)d_ws;

    k_zero_tables<<<128, 256, 0, stream>>>(ws);
    k_precompute<<<1, 512, 0, stream>>>(v_c1, v_c2, v_d, v_r, w_ih, W2a, ws);
    k_gru<<<4, 256, 0, stream>>>(d_seq, r_seq, Dtab, w_hh, b_ih, b_hh, ws,
                                 out + OUT_H);
    k_alpha<<<(BB * TT) / 16, 256, 0, stream>>>(out + OUT_H, W1a, b1a, W1b, b1b,
                                                out + OUT_ALPHA);
    k_cscan<<<1, 64, 0, stream>>>(c1_seq, c2_seq, d_seq, r_seq, Dtab,
                                  b2a, W2b, b2b, ws);
    {
        dim3 g1((NUM_C1 + 255) / 256, BB);
        k_fill_c1<<<g1, 256, 0, stream>>>(c1_seq, ws, out + OUT_C1);
        dim3 g2((NUM_C2 + 255) / 256, BB);
        k_fill_c2<<<g2, 256, 0, stream>>>(c2_seq, ws, out + OUT_C2);
    }
}